// DynamicTokenLearner_38835094290685
// MI455X (gfx1250) — compile-verified
//
#include <hip/hip_runtime.h>
#include <hip/hip_bf16.h>
#include <math.h>

// ---------------------------------------------------------------------------
// DynamicTokenLearner for gfx1250 (MI455X), fp32 end-to-end via
// V_WMMA_F32_16X16X4_F32.  B=1, N=65536, C=768, G=16, CG=48, T=64.
// ---------------------------------------------------------------------------

#define NTOK_SEQ   65536
#define CDIM       768
#define NGRP       16
#define CG_DIM     48
#define TMAX       64
#define LN_EPS     1e-5f

typedef float v2f __attribute__((ext_vector_type(2)));
typedef float v8f __attribute__((ext_vector_type(8)));

__device__ inline v8f v8f_zero() {
    v8f z = {0.f, 0.f, 0.f, 0.f, 0.f, 0.f, 0.f, 0.f};
    return z;
}

__device__ inline v8f wmma4(v2f a, v2f b, v8f c) {
    // D = A(16x4 f32) * B(4x16 f32) + C(16x16 f32)
    return __builtin_amdgcn_wmma_f32_16x16x4_f32(
        /*neg_a=*/false, a, /*neg_b=*/false, b,
        /*c_mod=*/(short)0, c, /*reuse_a=*/false, /*reuse_b=*/false);
}

// ---------------------- workspace layout (floats) --------------------------
#define WS_MU      0                         // [65536]
#define WS_RS      65536                     // [65536]
#define WS_POOL    131072                    // [768]
#define WS_RMAX    131840                    // [64]
#define WS_RSUM    131904                    // [64]
#define WS_NTOK    131968                    // [1] (int)
#define WS_OUTACC  132032                    // [64*768]

// ---------------------------------------------------------------------------
// K0: zero pooled + outacc
// ---------------------------------------------------------------------------
__global__ void k_init(float* ws) {
    int i = blockIdx.x * blockDim.x + threadIdx.x;
    if (i < CDIM)              ws[WS_POOL + i]   = 0.f;
    if (i < TMAX * CDIM)       ws[WS_OUTACC + i] = 0.f;
}

// ---------------------------------------------------------------------------
// K1: per-token LN stats (mu, rsigma) + pooled accumulation of (x-mu)*rs
//     grid 512 x 256 ; each wave handles 16 tokens
// ---------------------------------------------------------------------------
__global__ void k_ln_stats(const float* __restrict__ x, float* __restrict__ ws) {
    __shared__ float pl[CDIM];
    for (int c = threadIdx.x; c < CDIM; c += 256) pl[c] = 0.f;
    __syncthreads();

    const int wave = threadIdx.x >> 5;
    const int lane = threadIdx.x & 31;
    for (int i = 0; i < 16; ++i) {
        const int n = (blockIdx.x * 8 + wave) * 16 + i;
        const float* xr = x + (size_t)n * CDIM;
        float s = 0.f, s2 = 0.f;
        for (int c = lane; c < CDIM; c += 32) {
            float v = xr[c];
            s += v; s2 += v * v;
        }
        for (int m = 16; m > 0; m >>= 1) {
            s  += __shfl_xor(s,  m, 32);
            s2 += __shfl_xor(s2, m, 32);
        }
        const float mean = s * (1.f / (float)CDIM);
        const float var  = s2 * (1.f / (float)CDIM) - mean * mean;
        const float rsv  = rsqrtf(var + LN_EPS);
        if (lane == 0) {
            ws[WS_MU + n] = mean;
            ws[WS_RS + n] = rsv;
        }
        for (int c = lane; c < CDIM; c += 32)
            atomicAdd(&pl[c], (xr[c] - mean) * rsv);
    }
    __syncthreads();
    for (int c = threadIdx.x; c < CDIM; c += 256)
        atomicAdd(&ws[WS_POOL + c], pl[c]);
}

// ---------------------------------------------------------------------------
// K2: ratio = sigmoid(mean_n(xn) . pred_w + pred_b) ; num_tokens = floor(64*r)
// ---------------------------------------------------------------------------
__global__ void k_ratio(const float* __restrict__ tg, const float* __restrict__ tb,
                        const float* __restrict__ pw, const float* __restrict__ pb,
                        float* __restrict__ ws) {
    __shared__ float red[256];
    float s = 0.f;
    for (int c = threadIdx.x; c < CDIM; c += 256) {
        float mc = ws[WS_POOL + c] * (1.f / (float)NTOK_SEQ) * tg[c] + tb[c];
        s += mc * pw[c];
    }
    red[threadIdx.x] = s;
    __syncthreads();
    for (int st = 128; st > 0; st >>= 1) {
        if (threadIdx.x < st) red[threadIdx.x] += red[threadIdx.x + st];
        __syncthreads();
    }
    if (threadIdx.x == 0) {
        float r = 1.f / (1.f + __expf(-(red[0] + pb[0])));
        int nt = (int)floorf(r * (float)TMAX);
        nt = nt < 0 ? 0 : (nt > TMAX ? TMAX : nt);
        *(int*)(ws + WS_NTOK) = nt;
    }
}

// ---------------------------------------------------------------------------
// K3: scores[t][n] = sum_c relu(grouped(xn,w1))[n,c] * w2[t,c]
//     grid 256 x 256 ; each wave owns TWO 16-token tiles (32 rows) so every
//     w1/w2 B-fragment load feeds two WMMAs.  launch_bounds(256,2): block is
//     2 waves/SIMD, so allow the ~160 live VGPRs without scratch spills.
// ---------------------------------------------------------------------------
__global__ void __launch_bounds__(256, 2)
k_scores(const float* __restrict__ x,
         const float* __restrict__ tg, const float* __restrict__ tb,
         const float* __restrict__ w1, const float* __restrict__ w2,
         const float* __restrict__ ws,
         float* __restrict__ scores /* [64][65536] in d_out */) {
    __shared__ float hbuf[8][2 * 16 * CG_DIM];   // per-wave H tiles (2 x 16x48)
    const int wave = threadIdx.x >> 5;
    const int lane = threadIdx.x & 31;
    const int half = lane >> 4;
    const int lrow = lane & 15;
    const int n0   = (blockIdx.x * 8 + wave) * 32;
    float* hb = hbuf[wave];

    const int   row0 = n0 + lrow;
    const int   row1 = n0 + 16 + lrow;
    const float mu0 = ws[WS_MU + row0], rs0 = ws[WS_RS + row0];
    const float mu1 = ws[WS_MU + row1], rs1 = ws[WS_RS + row1];

    v8f S[2][4];
    for (int rt = 0; rt < 2; ++rt)
        for (int t = 0; t < 4; ++t) S[rt][t] = v8f_zero();

    for (int g = 0; g < NGRP; ++g) {
        const int cbase = g * CG_DIM;
        // --- A fragments: LN'd X for both row tiles ------------------------
        v2f A0[12], A1[12];
#pragma unroll
        for (int kk = 0; kk < 12; ++kk) {
            const int c0 = cbase + kk * 4 + half * 2;
            const float g0 = tg[c0], g1 = tg[c0 + 1];
            const float b0 = tb[c0], b1 = tb[c0 + 1];
            {
                const float* xp = x + (size_t)row0 * CDIM + c0;
                A0[kk].x = (xp[0] - mu0) * rs0 * g0 + b0;
                A0[kk].y = (xp[1] - mu0) * rs0 * g1 + b1;
            }
            {
                const float* xp = x + (size_t)row1 * CDIM + c0;
                A1[kk].x = (xp[0] - mu1) * rs1 * g0 + b0;
                A1[kk].y = (xp[1] - mu1) * rs1 * g1 + b1;
            }
        }
        // --- H = relu(Xg @ W1g^T): one B load feeds both row tiles ---------
#pragma unroll
        for (int nt = 0; nt < 3; ++nt) {
            v8f acc0 = v8f_zero(), acc1 = v8f_zero();
#pragma unroll
            for (int kk = 0; kk < 12; ++kk) {
                const int i0 = kk * 4 + half * 2;
                const float* wp = w1 + g * (CG_DIM * CG_DIM) + (nt * 16 + lrow) * CG_DIM + i0;
                v2f b; b.x = wp[0]; b.y = wp[1];
                acc0 = wmma4(A0[kk], b, acc0);
                acc1 = wmma4(A1[kk], b, acc1);
            }
#pragma unroll
            for (int r = 0; r < 8; ++r) {
                hb[(r + half * 8) * CG_DIM + nt * 16 + lrow]               = fmaxf(acc0[r], 0.f);
                hb[16 * CG_DIM + (r + half * 8) * CG_DIM + nt * 16 + lrow] = fmaxf(acc1[r], 0.f);
            }
        }
        asm volatile("s_wait_dscnt 0x0" ::: "memory");   // same-wave LDS RAW
        // --- S += Hg @ W2ᵀ slab: one w2 B load feeds both row tiles --------
#pragma unroll
        for (int kk = 0; kk < 12; ++kk) {
            const int i0 = kk * 4 + half * 2;
            v2f a0, a1;
            a0.x = hb[lrow * CG_DIM + i0];
            a0.y = hb[lrow * CG_DIM + i0 + 1];
            a1.x = hb[16 * CG_DIM + lrow * CG_DIM + i0];
            a1.y = hb[16 * CG_DIM + lrow * CG_DIM + i0 + 1];
#pragma unroll
            for (int tt = 0; tt < 4; ++tt) {
                const float* wp = w2 + (size_t)(tt * 16 + lrow) * CDIM + cbase + i0;
                v2f b; b.x = wp[0]; b.y = wp[1];
                S[0][tt] = wmma4(a0, b, S[0][tt]);
                S[1][tt] = wmma4(a1, b, S[1][tt]);
            }
        }
        asm volatile("s_wait_dscnt 0x0" ::: "memory");   // loads done before overwrite
    }
    // --- write raw scores: C-frag (M=n row, N=t col) -----------------------
#pragma unroll
    for (int rt = 0; rt < 2; ++rt)
#pragma unroll
        for (int tt = 0; tt < 4; ++tt)
#pragma unroll
            for (int r = 0; r < 8; ++r) {
                const int t = tt * 16 + lrow;
                const int n = n0 + rt * 16 + r + half * 8;
                scores[(size_t)t * NTOK_SEQ + n] = S[rt][tt][r];
            }
}

// ---------------------------------------------------------------------------
// K4: per-token-row softmax stats over N  (grid 64 x 256)
// ---------------------------------------------------------------------------
__global__ void k_softmax_stats(const float* __restrict__ scores, float* __restrict__ ws) {
    __shared__ float red[256];
    const int t = blockIdx.x;
    const float* srow = scores + (size_t)t * NTOK_SEQ;
    float m = -INFINITY;
    for (int n = threadIdx.x; n < NTOK_SEQ; n += 256) m = fmaxf(m, srow[n]);
    red[threadIdx.x] = m; __syncthreads();
    for (int st = 128; st > 0; st >>= 1) {
        if (threadIdx.x < st) red[threadIdx.x] = fmaxf(red[threadIdx.x], red[threadIdx.x + st]);
        __syncthreads();
    }
    const float rm = red[0];
    __syncthreads();
    float s = 0.f;
    for (int n = threadIdx.x; n < NTOK_SEQ; n += 256) s += __expf(srow[n] - rm);
    red[threadIdx.x] = s; __syncthreads();
    for (int st = 128; st > 0; st >>= 1) {
        if (threadIdx.x < st) red[threadIdx.x] += red[threadIdx.x + st];
        __syncthreads();
    }
    if (threadIdx.x == 0) { ws[WS_RMAX + t] = rm; ws[WS_RSUM + t] = red[0]; }
}

// ---------------------------------------------------------------------------
// K5: in-place masked softmax   sel = t<ntok ? exp(s-max)/sum : 0
// ---------------------------------------------------------------------------
__global__ void k_softmax_apply(float* __restrict__ scores, const float* __restrict__ ws) {
    const int idx  = blockIdx.x * blockDim.x + threadIdx.x;
    const int t    = idx >> 16;
    const int ntok = *(const int*)(ws + WS_NTOK);
    float v = 0.f;
    if (t < ntok)
        v = __expf(scores[idx] - ws[WS_RMAX + t]) / ws[WS_RSUM + t];
    scores[idx] = v;
}

// ---------------------------------------------------------------------------
// K6: out[t,c] += sum_n sel[t,n] * feat[n,c]   (split-K over n)
//     grid 512 x 512 (16 waves = 4 waves/SIMD; launch_bounds matches so the
//     acc[4][3] register tile never spills).  Each block: 8 chunks of 16 rows.
//     Phase 1: wave g computes feat group g into LDS (16x768, WMMA).
//     Phase 2: per wave, 4(m) x 3(c) register outer product per k-step:
//              4 sel A-frags + 3 feat B-frags feed 12 WMMAs.
// ---------------------------------------------------------------------------
__global__ void __launch_bounds__(512, 4)
k_outmm(const float* __restrict__ x,
        const float* __restrict__ tg, const float* __restrict__ tb,
        const float* __restrict__ wf,
        const float* __restrict__ sel,
        float* __restrict__ ws) {
    __shared__ float fb[16 * CDIM];            // feat tile 16 x 768 (48 KB)
    const int wave = threadIdx.x >> 5;         // 0..15
    const int lane = threadIdx.x & 31;
    const int half = lane >> 4;
    const int lrow = lane & 15;

    v8f acc[4][3];
    for (int m = 0; m < 4; ++m)
        for (int j = 0; j < 3; ++j) acc[m][j] = v8f_zero();

    for (int it = 0; it < 8; ++it) {
        const int n0 = (blockIdx.x * 8 + it) * 16;
        // ---- phase 1: feat[n0:n0+16][wave*48 : +48] -----------------------
        {
            const int g   = wave;
            const int row = n0 + lrow;
            const float muv = ws[WS_MU + row];
            const float rsv = ws[WS_RS + row];
            const int cbase = g * CG_DIM;
            v2f A[12];
#pragma unroll
            for (int kk = 0; kk < 12; ++kk) {
                const int c0 = cbase + kk * 4 + half * 2;
                const float* xp = x + (size_t)row * CDIM + c0;
                A[kk].x = (xp[0] - muv) * rsv * tg[c0]     + tb[c0];
                A[kk].y = (xp[1] - muv) * rsv * tg[c0 + 1] + tb[c0 + 1];
            }
#pragma unroll
            for (int nt = 0; nt < 3; ++nt) {
                v8f f = v8f_zero();
#pragma unroll
                for (int kk = 0; kk < 12; ++kk) {
                    const int i0 = kk * 4 + half * 2;
                    const float* wp = wf + g * (CG_DIM * CG_DIM) + (nt * 16 + lrow) * CG_DIM + i0;
                    v2f b; b.x = wp[0]; b.y = wp[1];
                    f = wmma4(A[kk], b, f);
                }
#pragma unroll
                for (int r = 0; r < 8; ++r)
                    fb[(r + half * 8) * CDIM + cbase + nt * 16 + lrow] = f[r];
            }
        }
        __syncthreads();
        // ---- phase 2: sel[64, n0:16] @ feat[16, 768] ----------------------
#pragma unroll
        for (int kk = 0; kk < 4; ++kk) {
            const int i0 = kk * 4 + half * 2;
            v2f av[4];
#pragma unroll
            for (int m = 0; m < 4; ++m) {
                const float* sp = sel + (size_t)(m * 16 + lrow) * NTOK_SEQ + n0 + i0;
                av[m].x = sp[0]; av[m].y = sp[1];
            }
            v2f bv[3];
#pragma unroll
            for (int j = 0; j < 3; ++j) {
                const int c0 = (wave * 3 + j) * 16;
                bv[j].x = fb[i0 * CDIM + c0 + lrow];
                bv[j].y = fb[(i0 + 1) * CDIM + c0 + lrow];
            }
#pragma unroll
            for (int m = 0; m < 4; ++m)
#pragma unroll
                for (int j = 0; j < 3; ++j)
                    acc[m][j] = wmma4(av[m], bv[j], acc[m][j]);
        }
        __syncthreads();
    }
    // ---- single atomic pass into out accumulator --------------------------
    for (int m = 0; m < 4; ++m)
        for (int j = 0; j < 3; ++j) {
            const int c = (wave * 3 + j) * 16 + lrow;
#pragma unroll
            for (int r = 0; r < 8; ++r) {
                const int t = m * 16 + r + half * 8;
                atomicAdd(&ws[WS_OUTACC + t * CDIM + c], acc[m][j][r]);
            }
        }
}

// ---------------------------------------------------------------------------
// K7: final LayerNorm of out rows + token mask  (grid 64 x 256)
// ---------------------------------------------------------------------------
__global__ void k_out_ln(const float* __restrict__ ws,
                         const float* __restrict__ ng, const float* __restrict__ nb,
                         float* __restrict__ out) {
    __shared__ float red[256];
    const int t    = blockIdx.x;
    const int ntok = *(const int*)(ws + WS_NTOK);
    const float* row = ws + WS_OUTACC + t * CDIM;

    float s = 0.f;
    for (int c = threadIdx.x; c < CDIM; c += 256) s += row[c];
    red[threadIdx.x] = s; __syncthreads();
    for (int st = 128; st > 0; st >>= 1) {
        if (threadIdx.x < st) red[threadIdx.x] += red[threadIdx.x + st];
        __syncthreads();
    }
    const float mean = red[0] * (1.f / (float)CDIM);
    __syncthreads();
    float s2 = 0.f;
    for (int c = threadIdx.x; c < CDIM; c += 256) {
        const float d = row[c] - mean;
        s2 += d * d;
    }
    red[threadIdx.x] = s2; __syncthreads();
    for (int st = 128; st > 0; st >>= 1) {
        if (threadIdx.x < st) red[threadIdx.x] += red[threadIdx.x + st];
        __syncthreads();
    }
    const float rsv = rsqrtf(red[0] * (1.f / (float)CDIM) + LN_EPS);
    for (int c = threadIdx.x; c < CDIM; c += 256) {
        float v = (t < ntok) ? ((row[c] - mean) * rsv * ng[c] + nb[c]) : 0.f;
        out[(size_t)t * CDIM + c] = v;
    }
}

// ---------------------------------------------------------------------------
extern "C" void kernel_launch(void* const* d_in, const int* in_sizes, int n_in,
                              void* d_out, int out_size, void* d_ws, size_t ws_size,
                              hipStream_t stream) {
    (void)in_sizes; (void)n_in; (void)out_size; (void)ws_size;
    const float* x   = (const float*)d_in[0];
    const float* tg  = (const float*)d_in[1];
    const float* tb  = (const float*)d_in[2];
    const float* w1  = (const float*)d_in[3];
    const float* w2  = (const float*)d_in[4];
    const float* wf  = (const float*)d_in[5];
    const float* pw  = (const float*)d_in[6];
    const float* pb  = (const float*)d_in[7];
    const float* ng  = (const float*)d_in[8];
    const float* nb  = (const float*)d_in[9];

    float* ws     = (float*)d_ws;
    float* sel    = (float*)d_out;                            // [64][65536]
    float* outptr = (float*)d_out + (size_t)TMAX * NTOK_SEQ;  // [64][768]

    k_init<<<(TMAX * CDIM + 255) / 256, 256, 0, stream>>>(ws);
    k_ln_stats<<<512, 256, 0, stream>>>(x, ws);
    k_ratio<<<1, 256, 0, stream>>>(tg, tb, pw, pb, ws);
    k_scores<<<256, 256, 0, stream>>>(x, tg, tb, w1, w2, ws, sel);
    k_softmax_stats<<<TMAX, 256, 0, stream>>>(sel, ws);
    k_softmax_apply<<<(TMAX * NTOK_SEQ) / 256, 256, 0, stream>>>(sel, ws);
    k_outmm<<<512, 512, 0, stream>>>(x, tg, tb, wf, sel, ws);
    k_out_ln<<<TMAX, 256, 0, stream>>>(ws, ng, nb, outptr);
}